// CNN_Entropy_19533511262736
// MI455X (gfx1250) — compile-verified
//
#include <hip/hip_runtime.h>
#include <math.h>

// Problem constants (fixed by the reference).
#define Bn 4
#define Cn 96
#define Hn 384
#define Wn 384
#define Kn 48                       // int(0.5 * 96)
#define NBINS 2048                  // key = center*128 + neigh8sum, max 15*128+120 = 2040
#define STRIP 32
#define SLOTS (STRIP + 2)
#define NSTRIPS (Hn / STRIP)        // 12
#define CHUNKS (SLOTS * (Wn / 4))   // 3264 float4 chunks per strip
#define CHUNK_ITERS 13              // ceil(3264/256): uniform per-wave async issue count
#define DENOMF 148996.0f            // (H+2)*(W+2)

// Low 32 bits of a generic pointer derived from LDS == LDS byte offset.
__device__ __forceinline__ uint32_t lds_offset(const void* p) {
  return (uint32_t)(uintptr_t)p;
}

// Stage one 34-row strip (rows clamped into [0,383]) into an LDS tile buffer.
// Exactly CHUNK_ITERS async-copy instructions per wave, every strip: the
// trailing partial iteration is clamped to a duplicate chunk (same src->same
// dst, benign) so s_wait_asynccnt can drain strips precisely.
__device__ __forceinline__ void stage_strip(const float* __restrict__ chan,
                                            float* __restrict__ buf,
                                            int firstRow, int tid) {
#pragma unroll
  for (int k = 0; k < CHUNK_ITERS; ++k) {
    int i = tid + (k << 8);
    i = (i < CHUNKS - 1) ? i : (CHUNKS - 1);
    const int s  = i / (Wn / 4);
    const int cx = (i % (Wn / 4)) * 4;
    int row = firstRow + s;
    row = (row < 0) ? 0 : ((row > Hn - 1) ? (Hn - 1) : row);
    const uint32_t loff  = lds_offset(&buf[s * Wn + cx]);
    const uint64_t gaddr = (uint64_t)(uintptr_t)(chan + (size_t)row * Wn + cx);
    asm volatile("global_load_async_to_lds_b128 %0, %1, off"
                 :: "v"(loff), "v"(gaddr) : "memory");
  }
}

// -------------------------------------------------------------------------
// Stage 1: per-channel 2048-bin histogram of (center*128 + 8-neighbor-sum),
// then entropy with p = count / 148996.  One workgroup per (b,c) channel.
// Double-buffered async global->LDS pipeline: strip N+1 streams into LDS
// while the 3x3 stencil + ds_add_u32 histogram runs on strip N.
// -------------------------------------------------------------------------
__global__ void __launch_bounds__(256)
entropy_kernel(const float* __restrict__ img, float* __restrict__ ent) {
  __shared__ float    tile[2][SLOTS * Wn];  // 2 x 52224 B
  __shared__ uint32_t hist[NBINS];          // 8192 B
  __shared__ float    red[256];             // 1024 B

  const int ch  = blockIdx.x;               // 0 .. B*C-1
  const int tid = threadIdx.x;
  const float* chan = img + (size_t)ch * (size_t)(Hn * Wn);

  for (int i = tid; i < NBINS; i += 256) hist[i] = 0u;

  // Prologue: start strip 0 into buffer 0 (overlaps the hist clear above).
  stage_strip(chan, tile[0], -1, tid);

  for (int strip = 0; strip < NSTRIPS; ++strip) {
    // Issue next strip's copies into the other buffer. That buffer was last
    // read by compute(strip-1), which all waves finished before the barrier
    // that ended the previous iteration.
    if (strip + 1 < NSTRIPS)
      stage_strip(chan, tile[(strip + 1) & 1], (strip + 1) * STRIP - 1, tid);

    // Drain the CURRENT strip's 13 copies (async loads complete in order;
    // at most the next strip's 13 remain outstanding afterwards).
    if (strip + 1 < NSTRIPS)
      asm volatile("s_wait_asynccnt 13" ::: "memory");
    else
      asm volatile("s_wait_asynccnt 0" ::: "memory");
    __syncthreads();   // all waves' copies for this strip visible in LDS

    // 3x3 stencil from LDS; top/bottom zero-padding handled by predicates
    // (staged halo rows are clamped, so exclude them at the image edges).
    const int y0 = strip * STRIP;
    const float* __restrict__ buf = tile[strip & 1];
    for (int i = tid; i < STRIP * Wn; i += 256) {
      const int ry = i / Wn;
      const int x  = i % Wn;
      const int y  = y0 + ry;
      const float* rm = &buf[(ry    ) * Wn];  // row y-1 (clamped copy)
      const float* rc = &buf[(ry + 1) * Wn];  // row y
      const float* rp = &buf[(ry + 2) * Wn];  // row y+1 (clamped copy)
      const float c = rc[x];
      float sum = c;
      if (x > 0)      sum += rc[x - 1];
      if (x < Wn - 1) sum += rc[x + 1];
      if (y > 0) {
        sum += rm[x];
        if (x > 0)      sum += rm[x - 1];
        if (x < Wn - 1) sum += rm[x + 1];
      }
      if (y < Hn - 1) {
        sum += rp[x];
        if (x > 0)      sum += rp[x - 1];
        if (x < Wn - 1) sum += rp[x + 1];
      }
      // sums of small integers are exact in f32; key in [0, 2040]
      const int key = ((int)c << 7) + (int)(sum - c);
      atomicAdd(&hist[key], 1u);
    }
    __syncthreads();   // compute done before the buffer is re-staged
  }

  // Entropy: sum over non-empty bins of -p*log2(p), p = count/148996.
  const float inv_denom = 1.0f / DENOMF;
  float h = 0.f;
  for (int b = tid; b < NBINS; b += 256) {
    const uint32_t cnt = hist[b];
    if (cnt) {
      const float p = (float)cnt * inv_denom;
      h -= p * log2f(p);
    }
  }
  red[tid] = h;
  __syncthreads();
  for (int s = 128; s > 0; s >>= 1) {
    if (tid < s) red[tid] += red[tid + s];
    __syncthreads();
  }
  if (tid == 0) ent[ch] = red[0];
}

// -------------------------------------------------------------------------
// Stage 2: top-48 of 96 channels per batch by rank-counting.
// rank_i = #{j: e_j > e_i} + #{j < i: e_j == e_i}  (== jax.lax.top_k order)
// -------------------------------------------------------------------------
__global__ void __launch_bounds__(96)
topk_kernel(const float* __restrict__ ent, int* __restrict__ idx) {
  const int b = blockIdx.x;
  const int i = threadIdx.x;          // 0..95
  __shared__ float e[Cn];
  e[i] = ent[b * Cn + i];
  __syncthreads();
  const float mine = e[i];
  int rank = 0;
  for (int j = 0; j < Cn; ++j) {
    const float ej = e[j];
    rank += (ej > mine) || (ej == mine && j < i);
  }
  if (rank < Kn) idx[b * Kn + rank] = i;
}

// -------------------------------------------------------------------------
// Stage 3: gather selected channels (pure bandwidth, float4 copies).
// -------------------------------------------------------------------------
__global__ void __launch_bounds__(256)
gather_kernel(const float* __restrict__ img, const int* __restrict__ idx,
              float4* __restrict__ out) {
  const uint32_t CH4 = (Hn * Wn) / 4;          // 36864 float4 per channel
  const uint32_t t = blockIdx.x * 256u + threadIdx.x;
  const uint32_t TOTAL4 = (uint32_t)Bn * Kn * CH4;
  if (t >= TOTAL4) return;
  const uint32_t b = t / (Kn * CH4);
  const uint32_t r = (t / CH4) % Kn;
  const uint32_t w = t % CH4;
  const int c = idx[b * Kn + r];
  const float4* src = (const float4*)img + ((size_t)b * Cn + (size_t)c) * CH4 + w;
  out[t] = *src;
}

// -------------------------------------------------------------------------
extern "C" void kernel_launch(void* const* d_in, const int* in_sizes, int n_in,
                              void* d_out, int out_size, void* d_ws, size_t ws_size,
                              hipStream_t stream) {
  (void)in_sizes; (void)n_in; (void)out_size; (void)ws_size;
  const float* img = (const float*)d_in[0];
  // d_in[1] is ratio = 0.5 (fixed by the reference; Kn = 48 baked in).

  float* ent = (float*)d_ws;                                   // 384 floats
  int*   idx = (int*)((char*)d_ws + 512 * sizeof(float));      // 192 ints

  entropy_kernel<<<Bn * Cn, 256, 0, stream>>>(img, ent);
  topk_kernel<<<Bn, Cn, 0, stream>>>(ent, idx);

  const uint32_t total4 = (uint32_t)Bn * Kn * (Hn * Wn / 4);   // 7,077,888
  gather_kernel<<<(total4 + 255) / 256, 256, 0, stream>>>(img, idx, (float4*)d_out);
}